// GeometricManyBodyProduct_13537736917625
// MI455X (gfx1250) — compile-verified
//
#include <hip/hip_runtime.h>
#include <utility>

// ---------------------------------------------------------------------------
// Compile-time Cl(3,0,1) tables (blade order matches the GATr reference)
// ---------------------------------------------------------------------------
struct Term { unsigned char i, j, k; signed char s; };
struct Tables {
  Term gp[192];          // geometric product nonzeros (metric 0,1,1,1)
  Term jn[84];           // join nonzeros (81 real + 3 zero padding)
  unsigned char i2m[16]; // blade index -> generator bitmask
};

constexpr int popc4(unsigned m) { int c = 0; for (int b = 0; b < 4; ++b) c += (m >> b) & 1u; return c; }
// Sign of reordering concat(sorted a, sorted b) into sorted order (inversion count).
constexpr int esign(unsigned a, unsigned b) {
  int sw = 0;
  for (int g = 0; g < 4; ++g) if ((b >> g) & 1u) sw += popc4(a >> (g + 1));
  return (sw & 1) ? -1 : 1;
}
constexpr unsigned I2M_[16] = {0,1,2,4,8,3,5,9,6,10,12,7,11,13,14,15};
constexpr int      M2I_[16] = {0,1,2,5,3,6,8,11,4,7,9,12,10,13,14,15};

constexpr Tables makeTables() {
  Tables T{};
  for (int i = 0; i < 16; ++i) T.i2m[i] = (unsigned char)I2M_[i];
  int n = 0;
  for (int j = 0; j < 16; ++j)
    for (int k = 0; k < 16; ++k) {
      unsigned a = I2M_[j], b = I2M_[k];
      if (a & b & 1u) continue;                    // e0*e0 = 0
      int s = esign(a, b);                         // e1..e3 square to +1
      unsigned r = a ^ b;
      T.gp[n] = Term{(unsigned char)M2I_[r], (unsigned char)j, (unsigned char)k, (signed char)s};
      ++n;                                         // ends at 192
    }
  int m = 0;
  for (int j = 0; j < 16; ++j)
    for (int k = 0; k < 16; ++k) {
      unsigned dj = (~I2M_[j]) & 0xFu, dk = (~I2M_[k]) & 0xFu;
      if (dj & dk) continue;                       // wedge of duals must be disjoint
      unsigned di = dj | dk;
      unsigned mi = (~di) & 0xFu;                  // result blade mask
      int s = esign(dj, dk);                       // wedge sign
      s *= esign(mi, di);                          // sd(i)
      s *= esign(I2M_[j], dj);                     // sd(j)
      s *= esign(I2M_[k], dk);                     // sd(k)
      T.jn[m] = Term{(unsigned char)M2I_[mi], (unsigned char)j, (unsigned char)k, (signed char)s};
      ++m;                                         // ends at 81
    }
  for (; m < 84; ++m) T.jn[m] = Term{0, 0, 0, 0};  // zero-weight padding
  return T;
}

constexpr Tables CT = makeTables();        // compile-time folded (template indexing)
__constant__ Tables DTAB = makeTables();   // device-resident (runtime indexing in setup)

// ---------------------------------------------------------------------------
// Geometry / layout constants
// ---------------------------------------------------------------------------
#define NPOS      131072      // B*H*S = 4*8*4096
#define NCH       8
#define KSTEPS    34          // 136 features / 4
#define NTILES    8           // 128 outputs / 16
// Workspace layout (floats)
#define WS_BL     0           // 17408: L B-fragments [kk][nt][lane][2]
#define WS_BR     17408       // 17408: R B-fragments
#define WS_WG     34816       // 8*192: sign-folded GP weights
#define WS_WJ     36352       // 8*88 : sign-folded join weights (padded, 16B-aligned rows)
// LDS layout (floats)
#define LDS_BFRAG 34816       // BL + BR staged fragments
#define LDS_XY    (LDS_BFRAG) // per-wave [16][128] X then Y, 4096 floats/wave
#define LDS_FLOATS (LDS_BFRAG + 8 * 4096)
#define TILES_PER_WAVE 4
#define MAIN_BLOCKS (NPOS / (16 * 8 * TILES_PER_WAVE))   // 256

typedef __attribute__((ext_vector_type(2))) float v2f;
typedef __attribute__((ext_vector_type(8))) float v8f;

// ---------------------------------------------------------------------------
// Setup: build WMMA-fragment-ordered equi-linear matrices + weight lists
// ---------------------------------------------------------------------------
// Laug[out=o*16+i][feat]: feat<128 -> c=feat>>4, k=feat&15:
//   k==i:                 w_mv[o,c,grade(i)]          (grade projections)
//   mask(i)=mask(k)|e0:   w_mv[o,c,5+grade(k)]        (e0-multiplication maps, sign +1)
// feat>=128 (scalars):    i==0 ? w_s[o,feat-128] : 0
__device__ __forceinline__ float equiEntry(const float* __restrict__ wmv,
                                           const float* __restrict__ wsc,
                                           int out, int feat) {
  const int o = out >> 4, i = out & 15;
  if (feat >= 128) return (i == 0) ? wsc[o * 8 + (feat - 128)] : 0.0f;
  const int cc = feat >> 4, kb = feat & 15;
  const unsigned mi = DTAB.i2m[i], mk = DTAB.i2m[kb];
  float v = 0.0f;
  if (kb == i) v += wmv[(o * 8 + cc) * 9 + __popc(mi)];
  if ((mi & 1u) && mk == (mi & ~1u)) v += wmv[(o * 8 + cc) * 9 + 5 + __popc(mk)];
  return v;
}

__global__ void __launch_bounds__(256)
gatr_setup(const float* __restrict__ lwmv, const float* __restrict__ lws,
           const float* __restrict__ rwmv, const float* __restrict__ rws,
           const float* __restrict__ gpw,  const float* __restrict__ jnw,
           float* __restrict__ ws) {
  const int idx = blockIdx.x * 256 + threadIdx.x;
  if (idx < 2 * 17408) {                       // B fragments, both sides
    const int side = idx / 17408;
    const int f    = idx % 17408;
    const int v    = f & 1;
    const int lane = (f >> 1) & 31;
    const int nt   = (f >> 6) & 7;
    const int kk   = f >> 9;
    const int kloc = ((lane >> 4) << 1) + v;   // ISA 16x16x4 B layout
    const int feat = kk * 4 + kloc;
    const int out  = nt * 16 + (lane & 15);
    const float val = equiEntry(side ? rwmv : lwmv, side ? rws : lws, out, feat);
    ws[(side ? WS_BR : WS_BL) + f] = val;
  } else if (idx < WS_WG + NCH * 192) {        // GP weights * Cayley sign
    const int t0 = idx - WS_WG;
    const int cc = t0 / 192, t = t0 % 192;
    const Term tm = DTAB.gp[t];
    ws[idx] = gpw[cc * 4096 + tm.i * 256 + tm.j * 16 + tm.k] * (float)tm.s;
  } else if (idx < WS_WJ + NCH * 88) {         // join weights * sign (padded)
    const int t0 = idx - WS_WJ;
    const int cc = t0 / 88, t = t0 % 88;
    float val = 0.0f;
    if (t < 84) {
      const Term tm = DTAB.jn[t];
      if (tm.s) val = jnw[cc * 4096 + tm.i * 256 + tm.j * 16 + tm.k] * (float)tm.s;
    }
    ws[idx] = val;
  }
}

// ---------------------------------------------------------------------------
// Unrolled sparse Cayley contraction (compile-time blade indices -> registers)
// ---------------------------------------------------------------------------
template <int T>
__device__ __forceinline__ void gp_term(float (&og)[16], const float (&x)[16],
                                        const float (&y)[16], float w) {
  og[(int)CT.gp[T].i] = fmaf(w * x[(int)CT.gp[T].j], y[(int)CT.gp[T].k], og[(int)CT.gp[T].i]);
}
template <int Q>
__device__ __forceinline__ void gp_chunk(float (&og)[16], const float (&x)[16],
                                         const float (&y)[16], const float4* __restrict__ wq) {
  const float4 w = wq[Q];
  gp_term<4 * Q + 0>(og, x, y, w.x);
  gp_term<4 * Q + 1>(og, x, y, w.y);
  gp_term<4 * Q + 2>(og, x, y, w.z);
  gp_term<4 * Q + 3>(og, x, y, w.w);
}
template <int... Qs>
__device__ __forceinline__ void gp_all(float (&og)[16], const float (&x)[16],
                                       const float (&y)[16], const float4* __restrict__ wq,
                                       std::integer_sequence<int, Qs...>) {
  (gp_chunk<Qs>(og, x, y, wq), ...);
}
template <int T>
__device__ __forceinline__ void jn_term(float (&oj)[16], const float (&x)[16],
                                        const float (&y)[16], float w) {
  oj[(int)CT.jn[T].i] = fmaf(w * x[(int)CT.jn[T].j], y[(int)CT.jn[T].k], oj[(int)CT.jn[T].i]);
}
template <int Q>
__device__ __forceinline__ void jn_chunk(float (&oj)[16], const float (&x)[16],
                                         const float (&y)[16], const float4* __restrict__ wq) {
  const float4 w = wq[Q];
  jn_term<4 * Q + 0>(oj, x, y, w.x);
  jn_term<4 * Q + 1>(oj, x, y, w.y);
  jn_term<4 * Q + 2>(oj, x, y, w.z);
  jn_term<4 * Q + 3>(oj, x, y, w.w);
}
template <int... Qs>
__device__ __forceinline__ void jn_all(float (&oj)[16], const float (&x)[16],
                                       const float (&y)[16], const float4* __restrict__ wq,
                                       std::integer_sequence<int, Qs...>) {
  (jn_chunk<Qs>(oj, x, y, wq), ...);
}

// ---------------------------------------------------------------------------
// Main fused kernel: WMMA equi-linear -> LDS -> sparse bilinear -> store
// ---------------------------------------------------------------------------
__global__ void __launch_bounds__(256)
gatr_main(const float* __restrict__ o_mv, const float* __restrict__ o_s,
          const float* __restrict__ refmv, const float* __restrict__ ws,
          float* __restrict__ outp) {
  extern __shared__ float lds[];
  const int tid = threadIdx.x;

  // Stage both B-fragment matrices (L2-resident, 139 KB) into LDS once/block.
  for (int i = tid; i < LDS_BFRAG; i += 256) lds[i] = ws[i];
  __syncthreads();

  const int wave = tid >> 5;
  const int lane = tid & 31;
  const int Mrow = lane & 15;            // A-fragment row (position within tile)
  const int hiK  = (lane >> 4) << 1;     // K sub-offset per ISA 16x16x4 A layout

  float* xw = lds + LDS_XY + wave * 4096;  // [16][128] X
  float* yw = xw + 2048;                   // [16][128] Y

  const int c    = lane & 7;             // channel handled in phase 2
  const int prow = lane >> 3;            // position sub-row
  const float4* wg = (const float4*)(ws + WS_WG + c * 192);
  const float4* wj = (const float4*)(ws + WS_WJ + c * 88);

  for (int tt = 0; tt < TILES_PER_WAVE; ++tt) {
    const int tile = (blockIdx.x * 8 + wave) * TILES_PER_WAVE + tt;
    const int pos0 = tile * 16;
    const float* rowA = o_mv + (size_t)(pos0 + Mrow) * 128;
    const float* rowS = o_s  + (size_t)(pos0 + Mrow) * 8;
    if (tt + 1 < TILES_PER_WAVE) __builtin_prefetch(rowA + 2048, 0, 1);

    // ---- Phase 1: X = Vaug * Laug^T, Y = Vaug * Raug^T via f32 WMMA ----
    const v8f vz = {0.f, 0.f, 0.f, 0.f, 0.f, 0.f, 0.f, 0.f};
    v8f ax[NTILES], ay[NTILES];
#pragma unroll
    for (int nt = 0; nt < NTILES; ++nt) { ax[nt] = vz; ay[nt] = vz; }

    // unroll 1: 16 independent WMMA chains already cover latency; deeper
    // unrolling doubles the B-fragment prefetch registers and pushes the
    // kernel over the 256-VGPR window (s_set_vgpr_msb churn between WMMAs).
#pragma unroll 1
    for (int kk = 0; kk < KSTEPS; ++kk) {
      v2f a;
      if (kk < 32) a = *(const v2f*)(rowA + (kk << 2) + hiK);        // mv features
      else         a = *(const v2f*)(rowS + ((kk - 32) << 2) + hiK); // scalar features
      const float* bB = lds + (kk << 9) + (lane << 1);
#pragma unroll
      for (int nt = 0; nt < NTILES; ++nt) {
        const v2f bL = *(const v2f*)(bB + nt * 64);
        const v2f bR = *(const v2f*)(bB + WS_BR + nt * 64);
        ax[nt] = __builtin_amdgcn_wmma_f32_16x16x4_f32(false, a, false, bL,
                                                       (short)0, ax[nt], false, false);
        ay[nt] = __builtin_amdgcn_wmma_f32_16x16x4_f32(false, a, false, bR,
                                                       (short)0, ay[nt], false, false);
      }
    }

    // ---- Scatter D fragments to per-wave LDS [pos][out] ----
    const int nlo   = lane & 15;
    const int rbase = (lane >> 4) << 3;
#pragma unroll
    for (int nt = 0; nt < NTILES; ++nt)
#pragma unroll
      for (int v = 0; v < 8; ++v) {
        xw[(rbase + v) * 128 + (nt << 4) + nlo] = ax[nt][v];
        yw[(rbase + v) * 128 + (nt << 4) + nlo] = ay[nt][v];
      }

    // ---- Phase 2: sparse Cayley bilinear per (position, channel) ----
#pragma unroll 1
    for (int it = 0; it < 4; ++it) {
      const int pos = (it << 2) + prow;
      float x[16], y[16];
      const float4* xv = (const float4*)(xw + pos * 128 + (c << 4));
      const float4* yv = (const float4*)(yw + pos * 128 + (c << 4));
#pragma unroll
      for (int q = 0; q < 4; ++q) {
        const float4 t = xv[q];
        x[4 * q + 0] = t.x; x[4 * q + 1] = t.y; x[4 * q + 2] = t.z; x[4 * q + 3] = t.w;
        const float4 u = yv[q];
        y[4 * q + 0] = u.x; y[4 * q + 1] = u.y; y[4 * q + 2] = u.z; y[4 * q + 3] = u.w;
      }
      float og[16], oj[16];
#pragma unroll
      for (int i = 0; i < 16; ++i) { og[i] = 0.0f; oj[i] = 0.0f; }
      gp_all(og, x, y, wg, std::make_integer_sequence<int, 48>{});
      jn_all(oj, x, y, wj, std::make_integer_sequence<int, 21>{});

      const float r15 = refmv[(size_t)(pos0 + pos) * 16 + 15];
      float4* op = (float4*)(outp + (size_t)(pos0 + pos) * 128 + (c << 4));
#pragma unroll
      for (int q = 0; q < 4; ++q) {
        float4 r;
        r.x = og[4 * q + 0] + r15 * oj[4 * q + 0] + y[4 * q + 0];
        r.y = og[4 * q + 1] + r15 * oj[4 * q + 1] + y[4 * q + 1];
        r.z = og[4 * q + 2] + r15 * oj[4 * q + 2] + y[4 * q + 2];
        r.w = og[4 * q + 3] + r15 * oj[4 * q + 3] + y[4 * q + 3];
        op[q] = r;
      }
    }
  }
}

// ---------------------------------------------------------------------------
extern "C" void kernel_launch(void* const* d_in, const int* in_sizes, int n_in,
                              void* d_out, int out_size, void* d_ws, size_t ws_size,
                              hipStream_t stream) {
  (void)in_sizes; (void)n_in; (void)out_size; (void)ws_size;
  const float* o_mv = (const float*)d_in[0];
  const float* o_s  = (const float*)d_in[1];
  const float* ref  = (const float*)d_in[2];
  const float* lwmv = (const float*)d_in[3];
  const float* lws  = (const float*)d_in[4];
  const float* rwmv = (const float*)d_in[5];
  const float* rws  = (const float*)d_in[6];
  const float* gpw  = (const float*)d_in[7];
  const float* jnw  = (const float*)d_in[8];
  float* out = (float*)d_out;
  float* ws  = (float*)d_ws;

  const int setupElems = WS_WJ + NCH * 88;               // 37056
  gatr_setup<<<(setupElems + 255) / 256, 256, 0, stream>>>(lwmv, lws, rwmv, rws, gpw, jnw, ws);

  const size_t ldsBytes = (size_t)LDS_FLOATS * sizeof(float);  // 270336 B (<320KB/WGP)
  gatr_main<<<MAIN_BLOCKS, 256, ldsBytes, stream>>>(o_mv, o_s, ref, ws, out);
}